// Linear_56169582297408
// MI455X (gfx1250) — compile-verified
//
#include <hip/hip_runtime.h>
#include <stdint.h>

typedef __attribute__((ext_vector_type(2))) float v2f;
typedef __attribute__((ext_vector_type(8))) float v8f;

#define DIN  8
#define DOUT 8
#define EDGES_PER_WAVE 32
#define WAVES_PER_BLOCK 8
#define BLOCK_THREADS 256

// Per-edge y = W[widx[e]] * x[iidx[e]] using V_WMMA_F32_16X16X4_F32.
// Two edges per 16x16 WMMA tile (block-diagonal), K=8 via two chained WMMAs.
//
// A (16x4, fp32) layout (ISA 7.12.2): lanes 0-15 hold M=lane with
//   VGPR0=K0, VGPR1=K1; lanes 16-31 hold M=lane-16 with VGPR0=K2, VGPR1=K3.
// B (4x16) mirrors with N instead of M.
// C/D (16x16): VGPR v: lanes 0-15 -> D[v][lane], lanes 16-31 -> D[v+8][lane-16].
//   => y(e0)[o] = D[o][0]   = acc[o] in lane 0
//      y(e1)[o] = D[o+8][8] = acc[o] in lane 24
__global__ __launch_bounds__(BLOCK_THREADS)
void edge_gemv_wmma(const float* __restrict__ values,
                    const float* __restrict__ weights,
                    const int*   __restrict__ input_idx,
                    const int*   __restrict__ weight_idx,
                    float*       __restrict__ out,
                    int E)
{
    const int  lane = threadIdx.x & 31;
    const int  wave = threadIdx.x >> 5;
    const long base = ((long)blockIdx.x * WAVES_PER_BLOCK + wave) * EDGES_PER_WAVE;
    if (base >= (long)E) return;

    // Coalesced index fetch: lane L owns edge (base + L).
    const int my_iidx = input_idx[base + lane];
    const int my_widx = weight_idx[base + lane];

    // Static lane roles for the WMMA operand layouts.
    const int m   = lane & 15;        // A-row / B-col within 16
    const int sel = (m >> 3) & 1;     // 0 -> edge0 half, 1 -> edge1 half
    const int row = m & 7;            // W row
    const int kk  = (lane >> 4) * 2;  // K sub-offset {0,2} within a K=4 chunk

    // Loop-invariant per-lane byte offsets; gathers become saddr + u32 voffset.
    const uint32_t wconst = (uint32_t)(row * DIN + kk) * 4u; // within one 256B W
    const uint32_t xconst = (uint32_t)kk * 4u;               // within one 32B x
    const char* wbase = (const char*)weights;
    const char* xbase = (const char*)values;

    // Store lanes: lane 0 -> even edge, lane 24 -> odd edge. Per-pair offset is
    // a compile-time immediate after full unroll.
    float* obase = out + (size_t)base * DOUT + (lane == 24 ? DOUT : 0);
    const bool store_lane = (lane == 0) || (lane == 24);

#pragma unroll
    for (int p = 0; p < EDGES_PER_WAVE / 2; ++p) {
        // Constant-lane broadcasts: v_readlane_b32 -> SGPR, no LDS traffic.
        const int w0 = __builtin_amdgcn_readlane(my_widx, 2 * p);
        const int w1 = __builtin_amdgcn_readlane(my_widx, 2 * p + 1);
        const int i0 = __builtin_amdgcn_readlane(my_iidx, 2 * p);
        const int i1 = __builtin_amdgcn_readlane(my_iidx, 2 * p + 1);

        const uint32_t woff = ((uint32_t)(sel ? w1 : w0) << 8) + wconst; // *256B
        const uint32_t xoff = ((uint32_t)(sel ? i1 : i0) << 5) + xconst; // *32B

        // A: 32 lanes together read exactly the two 8x8 matrices (512 B).
        const v2f a_lo = *(const v2f*)(wbase + woff);        // K = kk, kk+1
        const v2f a_hi = *(const v2f*)(wbase + woff + 16);   // K = 4+kk, 5+kk
        // B: column n carries x of its edge (replicated -> L0 hits).
        const v2f b_lo = *(const v2f*)(xbase + xoff);
        const v2f b_hi = *(const v2f*)(xbase + xoff + 16);

        v8f acc = {};
        acc = __builtin_amdgcn_wmma_f32_16x16x4_f32(
                  false, a_lo, false, b_lo, (short)0, acc, false, false);
        acc = __builtin_amdgcn_wmma_f32_16x16x4_f32(
                  false, a_hi, false, b_hi, (short)0, acc, false, false);

        // 32 B store per active lane; consecutive edges -> contiguous 64 B.
        if (store_lane) {
            *(v8f*)(obase + p * 2 * DOUT) = acc;
        }
    }
}

extern "C" void kernel_launch(void* const* d_in, const int* in_sizes, int n_in,
                              void* d_out, int out_size, void* d_ws, size_t ws_size,
                              hipStream_t stream) {
    const float* values     = (const float*)d_in[0];  // [NUM_VALUES, 8]
    const float* weights    = (const float*)d_in[1];  // [NUM_WEIGHTS, 8, 8]
    const int*   input_idx  = (const int*)d_in[2];    // [E]
    const int*   weight_idx = (const int*)d_in[3];    // [E]
    float*       out        = (float*)d_out;          // [E, 8]

    const int E = in_sizes[2];
    const int edges_per_block = WAVES_PER_BLOCK * EDGES_PER_WAVE; // 256
    const int blocks = (E + edges_per_block - 1) / edges_per_block;

    edge_gemv_wmma<<<blocks, BLOCK_THREADS, 0, stream>>>(
        values, weights, input_idx, weight_idx, out, E);
}